// Preprocessor_25890062860635
// MI455X (gfx1250) — compile-verified
//
#include <hip/hip_runtime.h>

#define Z 16
#define VD 100
#define TLEN 10000
#define NTRIALS 64
#define JITTER 1e-6f

typedef float v2f __attribute__((ext_vector_type(2)));
typedef float v4f __attribute__((ext_vector_type(4)));
typedef float v8f __attribute__((ext_vector_type(8)));

// ---------------------------------------------------------------------------
// V_WMMA_F32_16X16X4_F32 wrapper.  D(16x16,f32) = A(16x4,f32) x B(4x16,f32) + C
// Layouts (CDNA5 ISA 7.12.2, wave32):
//   A: lane l: m = l&15 ; .x = A[m][k0], .y = A[m][k0+1], k0=(l>>4)*2
//   B: lane l: n = l&15 ; .x = B[k0][n], .y = B[k0+1][n]
//   C/D: vgpr r: lanes 0-15 -> M=r, lanes 16-31 -> M=r+8 ; N = lane&15
// ---------------------------------------------------------------------------
static __device__ __forceinline__ v8f wmma4(v2f a, v2f b, v8f c) {
  return __builtin_amdgcn_wmma_f32_16x16x4_f32(
      /*neg_a=*/false, a, /*neg_b=*/false, b,
      /*c_mod=*/(short)0, c, /*reuse_a=*/false, /*reuse_b=*/false);
}

static __device__ __forceinline__ v8f zero8() {
  v8f z = {0.f, 0.f, 0.f, 0.f, 0.f, 0.f, 0.f, 0.f};
  return z;
}

static __device__ __forceinline__ v2f frag_a(const float* p, int stride, int kb, int lane) {
  int m = lane & 15, k0 = ((lane >> 4) << 1) + kb;
  v2f a;
  a.x = p[m * stride + k0];
  a.y = p[m * stride + k0 + 1];
  return a;
}
static __device__ __forceinline__ v2f frag_b(const float* p, int stride, int kb, int n0, int lane) {
  int n = (lane & 15) + n0, k0 = ((lane >> 4) << 1) + kb;
  v2f b;
  b.x = p[k0 * stride + n];
  b.y = p[(k0 + 1) * stride + n];
  return b;
}
// logical B = P^T for row-major P  (B[k][n] = P[n][k])
static __device__ __forceinline__ v2f frag_bt(const float* p, int stride, int kb, int n0, int lane) {
  int n = (lane & 15) + n0, k0 = ((lane >> 4) << 1) + kb;
  v2f b;
  b.x = p[n * stride + k0];
  b.y = p[n * stride + k0 + 1];
  return b;
}

// v_rcp_f32 + one Newton step (full f32 accuracy, avoids v_div_* sequence)
static __device__ __forceinline__ float fast_rcp(float x) {
  float r = __builtin_amdgcn_rcpf(x);
  float e = fmaf(-x, r, 1.0f);
  return fmaf(r, e, r);
}
// broadcast lane `src` (compile-time) -> v_readlane_b32, uniform result
static __device__ __forceinline__ float bcast(float x, int src) {
  return __int_as_float(__builtin_amdgcn_readlane(__float_as_int(x), src));
}

// ===========================================================================
// Kernel 1: covariance recursion (serial over T).  One workgroup, 8 waves.
// Push-through form (R factored ONCE):
//   G  = I + Sigma*M                  (WMMA, wave 0)    M = W^T R^-1 W
//   Sf = G^-1 * Sigma                 (register Gauss-Jordan, wave 0 only:
//                                      column-per-lane, v_readlane broadcast,
//                                      fully unrolled, NO barriers, NO LDS)
//   K  = Sf * (W^T R^-1)              (WMMA, waves 0..6, 7 column tiles)
//   T1 = A*Sf (wave 7) ; Sigma' = T1*A^T + Q (wave 0)
// Only 2 block barriers per step.
// ===========================================================================
__global__ __launch_bounds__(256, 1) void kal_cov(
    const float* __restrict__ Ain, const float* __restrict__ Bin,
    const float* __restrict__ Win, const float* __restrict__ S0h,
    const float* __restrict__ Rh,
    float* __restrict__ SigOut,   // (T,16,16)
    float* __restrict__ KOut)     // (T,16,100)
{
  const int SR = 101;
  __shared__ float Rm[100 * 101];     // R -> Cholesky factor (one-time)
  __shared__ float Xc[100 * 16];      // X = R^-1 W
  __shared__ float WtRi[16 * 113];    // W^T R^-1 zero-padded to 16x112
  __shared__ float Mm[16 * 17];
  __shared__ float Als[16 * 17];
  __shared__ float Qls[16 * 17];
  __shared__ float Sg[16 * 17];       // Sigma carry
  __shared__ float T1[16 * 17];
  __shared__ float Gls[16 * 17];      // G staging (wave0 private use)
  __shared__ float SfLs[16 * 17];     // Sf broadcast to all waves

  const int tid = threadIdx.x;
  const int lane = tid & 31;
  const int wv = tid >> 5;

  // ---- P1: R = tril(Rh) tril(Rh)^T + jitter*I ; Q, Sigma0, A ----
  for (int idx = tid; idx < 100 * 100; idx += 256) {
    int i = idx / 100, j = idx % 100;
    int kmax = (i < j) ? i : j;
    float acc = 0.f;
    for (int k = 0; k <= kmax; ++k) acc += Rh[i * 100 + k] * Rh[j * 100 + k];
    if (i == j) acc += JITTER;
    Rm[i * SR + j] = acc;
  }
  {
    int i = tid >> 4, j = tid & 15;
    int kmax = (i < j) ? i : j;
    float q = 0.f, s0 = 0.f;
    for (int k = 0; k <= kmax; ++k) {
      q += Bin[i * 16 + k] * Bin[j * 16 + k];
      s0 += S0h[i * 16 + k] * S0h[j * 16 + k];
    }
    if (i == j) { q += JITTER; s0 += JITTER; }
    Qls[i * 17 + j] = q;
    Sg[i * 17 + j] = s0;
    Als[i * 17 + j] = Ain[i * 16 + j];
  }

  // ---- P2: one-time 100x100 Cholesky in LDS ----
  for (int k = 0; k < 100; ++k) {
    __syncthreads();
    float dkk = Rm[k * SR + k];
    float s = sqrtf(dkk);
    float rinv = 1.f / s;
    __syncthreads();
    if (tid == 0) Rm[k * SR + k] = s;
    for (int i = k + 1 + tid; i < 100; i += 256) Rm[i * SR + k] *= rinv;
    __syncthreads();
    int n = 99 - k;
    for (int idx = tid; idx < n * n; idx += 256) {
      int i = k + 1 + idx / n, j = k + 1 + idx % n;
      Rm[i * SR + j] -= Rm[i * SR + k] * Rm[j * SR + k];
    }
  }
  __syncthreads();

  // ---- P3: X = R^-1 W (triangular solves, one column per thread) ----
  if (tid < 16) {
    for (int i = 0; i < 100; ++i) {
      float acc = Win[i * 16 + tid];
      for (int j = 0; j < i; ++j) acc -= Rm[i * SR + j] * Xc[j * 16 + tid];
      Xc[i * 16 + tid] = acc / Rm[i * SR + i];
    }
    for (int i = 99; i >= 0; --i) {
      float acc = Xc[i * 16 + tid];
      for (int j = i + 1; j < 100; ++j) acc -= Rm[j * SR + i] * Xc[j * 16 + tid];
      Xc[i * 16 + tid] = acc / Rm[i * SR + i];
    }
  }
  __syncthreads();

  // ---- P4: WtRi = X^T (padded), M = W^T X ----
  for (int idx = tid; idx < 16 * 112; idx += 256) {
    int z = idx / 112, d = idx % 112;
    WtRi[z * 113 + d] = (d < 100) ? Xc[d * 16 + z] : 0.f;
  }
  {
    int m = tid >> 4, n2 = tid & 15;
    float acc = 0.f;
    for (int d = 0; d < 100; ++d) acc += Win[d * 16 + m] * Xc[d * 16 + n2];
    Mm[m * 17 + n2] = acc;
  }
  __syncthreads();

  // ---- serial covariance loop: 2 barriers / step ----
  for (int t = 0; t < TLEN; ++t) {
    if (wv == 0) {
      // Stage G: G = I + Sg*Mm  (WMMA) -> Gls
      int hi = lane >> 4, col = lane & 15;
      {
        v8f acc;
#pragma unroll
        for (int rr = 0; rr < 8; ++rr) acc[rr] = ((rr + 8 * hi) == col) ? 1.f : 0.f;
#pragma unroll
        for (int kb = 0; kb < 16; kb += 4)
          acc = wmma4(frag_a(Sg, 17, kb, lane), frag_b(Mm, 17, kb, 0, lane), acc);
#pragma unroll
        for (int rr = 0; rr < 8; ++rr) Gls[(rr + 8 * hi) * 17 + col] = acc[rr];
      }
      // Register Gauss-Jordan on [G | Sigma], column-per-lane:
      //   lanes 0..15 hold G columns, lanes 16..31 hold Sigma columns.
      float colv[16];
      {
        const float* base = (lane < 16) ? (Gls + lane) : (Sg + (lane - 16));
#pragma unroll
        for (int r = 0; r < 16; ++r) colv[r] = base[r * 17];
      }
#pragma unroll
      for (int p = 0; p < 16; ++p) {
        float fp[16];
#pragma unroll
        for (int r = 0; r < 16; ++r) fp[r] = bcast(colv[r], p);  // G column p
        float ip = fast_rcp(fp[p]);
        colv[p] *= ip;  // scale pivot row
#pragma unroll
        for (int r = 0; r < 16; ++r)
          if (r != p) colv[r] = fmaf(-fp[r], colv[p], colv[r]);
      }
      if (lane >= 16) {  // lanes 16..31 hold Sf = G^-1 Sigma columns
#pragma unroll
        for (int r = 0; r < 16; ++r) SfLs[r * 17 + (lane - 16)] = colv[r];
      }
    }
    __syncthreads();  // B1: Sf visible to all waves

    // Stage K: K = Sf * WtRi (waves 0..6, 7 column tiles); wave 7: T1 = A*Sf
    if (wv < 7) {
      int n0 = wv * 16;
      v8f acc = zero8();
#pragma unroll
      for (int kb = 0; kb < 16; kb += 4)
        acc = wmma4(frag_a(SfLs, 17, kb, lane), frag_b(WtRi, 113, kb, n0, lane), acc);
      int hi = lane >> 4, col = lane & 15, d = n0 + col;
      if (d < 100) {
#pragma unroll
        for (int rr = 0; rr < 8; ++rr)
          KOut[(size_t)t * (Z * VD) + (rr + 8 * hi) * VD + d] = acc[rr];
      }
    } else {
      v8f acc = zero8();
#pragma unroll
      for (int kb = 0; kb < 16; kb += 4)
        acc = wmma4(frag_a(Als, 17, kb, lane), frag_b(SfLs, 17, kb, 0, lane), acc);
      int hi = lane >> 4, col = lane & 15;
#pragma unroll
      for (int rr = 0; rr < 8; ++rr) T1[(rr + 8 * hi) * 17 + col] = acc[rr];
    }
    __syncthreads();  // B2: T1 ready, SfLs consumed

    // Stage Sigma': Sigma = T1*A^T + Q (wave 0); emit Sigmas_diffused[t]
    if (wv == 0) {
      int hi = lane >> 4, col = lane & 15;
      v8f acc;
#pragma unroll
      for (int rr = 0; rr < 8; ++rr) acc[rr] = Qls[(rr + 8 * hi) * 17 + col];
#pragma unroll
      for (int kb = 0; kb < 16; kb += 4)
        acc = wmma4(frag_a(T1, 17, kb, lane), frag_bt(Als, 17, kb, 0, lane), acc);
#pragma unroll
      for (int rr = 0; rr < 8; ++rr) {
        int row = rr + 8 * hi;
        Sg[row * 17 + col] = acc[rr];  // read next iter by wave0 only
        __builtin_nontemporal_store(acc[rr],
            &SigOut[(size_t)t * (Z * Z) + row * Z + col]);
      }
    }
    // no barrier needed: Sg/Gls/SfLs/T1 producers+consumers ordered by B1/B2
  }
}

// ===========================================================================
// Kernel 2: mean recursion.  4 workgroups x 16 trials, 2 barriers / step.
//   stage1: innov = v_t - mu*W^T  (waves 0..6, 7 col tiles) | wave 7 stages K_t
//   stage2: wave 0: mu_f = mu + innov*K_t^T  (28 back-to-back WMMA accumulates)
//           then mu' = mu_f*A^T -> mus_diffused[t]
// v (trials,vd,T): 4-step chunks staged via non-temporal 16B loads along
// the contiguous t axis; Sigma/mu outputs stored non-temporally so L2 keeps K.
// ===========================================================================
__global__ __launch_bounds__(256, 1) void kal_mean(
    const float* __restrict__ vin, const float* __restrict__ Ain,
    const float* __restrict__ Win, const float* __restrict__ mu0,
    const float* __restrict__ Kg,     // (T,16,100) from kal_cov
    float* __restrict__ musOut)       // (T,64,16)
{
  __shared__ float Wt[16 * 113];      // W^T zero-padded to 16x112
  __shared__ float Kt[16 * 113];      // K_t staged, zero-padded
  __shared__ float innv[16 * 113];    // innovation, zero-padded
  __shared__ float mu[16 * 17];
  __shared__ float muf[16 * 17];
  __shared__ float Als[16 * 17];
  __shared__ __align__(16) float vbuf[16 * 100 * 4];

  const int tid = threadIdx.x, lane = tid & 31, wv = tid >> 5;
  const int tr0 = blockIdx.x * 16;

  for (int idx = tid; idx < 16 * 112; idx += 256) {
    int z = idx / 112, d = idx % 112;
    Wt[z * 113 + d] = (d < 100) ? Win[d * 16 + z] : 0.f;
  }
  {
    int r = tid >> 4, c = tid & 15;
    Als[r * 17 + c] = Ain[r * 16 + c];
    mu[r * 17 + c] = mu0[c];
  }
  __syncthreads();

  for (int t = 0; t < TLEN; ++t) {
    int tc = t & 3;
    if (tc == 0) {  // stage 4-step chunk of v (streamed once -> non-temporal)
      for (int idx = tid; idx < 1600; idx += 256) {
        int tr = idx / 100, d = idx % 100;
        const v4f q = __builtin_nontemporal_load(
            reinterpret_cast<const v4f*>(
                &vin[((size_t)(tr0 + tr) * VD + d) * TLEN + t]));
        *reinterpret_cast<v4f*>(&vbuf[idx * 4]) = q;
      }
      __syncthreads();
    }

    // Stage 1: innov tiles on waves 0..6; wave 7 stages K_t + prefetch K_{t+1}
    if (wv < 7) {
      int n0 = wv * 16;
      v8f acc = zero8();
#pragma unroll
      for (int kb = 0; kb < 16; kb += 4)
        acc = wmma4(frag_a(mu, 17, kb, lane), frag_b(Wt, 113, kb, n0, lane), acc);
      int hi = lane >> 4, col = lane & 15, d = n0 + col;
#pragma unroll
      for (int rr = 0; rr < 8; ++rr) {
        int row = rr + 8 * hi;
        innv[row * 113 + d] =
            (d < 100) ? (vbuf[(row * 100 + d) * 4 + tc] - acc[rr]) : 0.f;
      }
    } else {
      for (int idx = lane; idx < 16 * 112; idx += 32) {
        int z = idx / 112, d = idx % 112;
        Kt[z * 113 + d] = (d < 100) ? Kg[(size_t)t * (Z * VD) + z * VD + d] : 0.f;
      }
      if (t + 1 < TLEN) {
        for (int off = lane * 16; off < Z * VD; off += 512)
          __builtin_prefetch(&Kg[(size_t)(t + 1) * (Z * VD) + off], 0, 1);
      }
    }
    __syncthreads();  // B1: innv + Kt ready

    // Stage 2 (wave 0): mu_f = mu + innov*K_t^T, then mu' = mu_f*A^T
    if (wv == 0) {
      int hi = lane >> 4, col = lane & 15;
      v8f acc;
#pragma unroll
      for (int rr = 0; rr < 8; ++rr) acc[rr] = mu[(rr + 8 * hi) * 17 + col];
#pragma unroll
      for (int cb = 0; cb < 28; ++cb) {  // K = 112 in 28 chained accumulates
        int kb = cb * 4;
        acc = wmma4(frag_a(innv, 113, kb, lane), frag_bt(Kt, 113, kb, 0, lane), acc);
      }
#pragma unroll
      for (int rr = 0; rr < 8; ++rr) muf[(rr + 8 * hi) * 17 + col] = acc[rr];
      // same-wave LDS ordering: muf store -> frag_a load is in-order
      v8f acc2 = zero8();
#pragma unroll
      for (int kb = 0; kb < 16; kb += 4)
        acc2 = wmma4(frag_a(muf, 17, kb, lane), frag_bt(Als, 17, kb, 0, lane), acc2);
#pragma unroll
      for (int rr = 0; rr < 8; ++rr) {
        int row = rr + 8 * hi;
        mu[row * 17 + col] = acc2[rr];
        __builtin_nontemporal_store(acc2[rr],
            &musOut[(size_t)t * (NTRIALS * Z) + (size_t)(tr0 + row) * Z + col]);
      }
    }
    __syncthreads();  // B2: mu updated for next step
  }
}

extern "C" void kernel_launch(void* const* d_in, const int* in_sizes, int n_in,
                              void* d_out, int out_size, void* d_ws, size_t ws_size,
                              hipStream_t stream) {
  (void)in_sizes; (void)n_in; (void)d_ws; (void)ws_size; (void)out_size;
  const float* v   = (const float*)d_in[0];
  const float* A   = (const float*)d_in[1];
  const float* B   = (const float*)d_in[2];
  const float* W   = (const float*)d_in[3];
  const float* mu0 = (const float*)d_in[4];
  const float* S0h = (const float*)d_in[5];
  const float* Rh  = (const float*)d_in[6];

  float* out = (float*)d_out;
  float* Sig = out;                                                 // T*16*16
  float* Ks  = out + (size_t)TLEN * Z * Z;                          // T*16*100
  float* mus = out + (size_t)TLEN * Z * Z + (size_t)TLEN * Z * VD;  // T*64*16

  hipLaunchKernelGGL(kal_cov, dim3(1), dim3(256), 0, stream,
                     A, B, W, S0h, Rh, Sig, Ks);
  hipLaunchKernelGGL(kal_mean, dim3(NTRIALS / 16), dim3(256), 0, stream,
                     v, A, W, mu0, Ks, mus);
}